// RGCNLayer_85942295593555
// MI455X (gfx1250) — compile-verified
//
#include <hip/hip_runtime.h>

#define NN   50000
#define EE   800000
#define RR   6
#define FIN  128
#define FOUT 128

typedef __attribute__((ext_vector_type(2))) float v2f;
typedef __attribute__((ext_vector_type(8))) float v8f;

// ---- workspace layout (in floats) ----
// [deg | S] zeroed every call; S doubles as the scatter target AND (in-place)
// the normalized per-relation output op[r,n,:]. u = [N,256] attention output.
static const size_t DEG_OFF = 0;
static const size_t DEG_SZ  = (size_t)RR * NN;                  //   300,000
static const size_t S_OFF   = DEG_OFF + DEG_SZ;
static const size_t S_SZ    = (size_t)RR * NN * FOUT;           // 38,400,000
static const size_t U_OFF   = S_OFF + S_SZ;                     // u: [N, 256]

// ---------------- zero fill (deg + S) ----------------
__global__ void k_zero(float4* __restrict__ p, long n4) {
    long i      = (long)blockIdx.x * blockDim.x + threadIdx.x;
    long stride = (long)gridDim.x * blockDim.x;
    float4 z = make_float4(0.f, 0.f, 0.f, 0.f);
    for (; i < n4; i += stride) p[i] = z;
}

// ---------------- per-(rel,dst) degree ----------------
__global__ void k_deg(const int* __restrict__ erel, const int* __restrict__ edst,
                      float* __restrict__ deg) {
    int e = blockIdx.x * blockDim.x + threadIdx.x;
    if (e >= EE) return;
    unsafeAtomicAdd(&deg[(size_t)erel[e] * NN + edst[e]], 1.0f);
}

// ---------------- edge scatter: S[rel,dst,:] += node_h[src,:] ----------------
// one wave per edge, 4 floats per lane; gather table node_h is 25.6 MB -> L2-hot
__global__ void k_scatter(const float* __restrict__ node_h, const int* __restrict__ esrc,
                          const int* __restrict__ edst, const int* __restrict__ erel,
                          float* __restrict__ S) {
    long t = (long)blockIdx.x * blockDim.x + threadIdx.x;
    int  e = (int)(t >> 5);
    if (e >= EE) return;
    int c = ((int)t & 31) * 4;
    int rl = erel[e], s = esrc[e], d = edst[e];
    float4 m = *(const float4*)(node_h + (size_t)s * FIN + c);
    float* ap = S + ((size_t)rl * NN + d) * FOUT + c;
    unsafeAtomicAdd(ap + 0, m.x);
    unsafeAtomicAdd(ap + 1, m.y);
    unsafeAtomicAdd(ap + 2, m.z);
    unsafeAtomicAdd(ap + 3, m.w);
}

// -------- GEMM1 (in place): op[r,n,:] = (S[r,n,:] / (1+deg[r,n])) @ W[r] --------
// FP32 WMMA 16x16x4. block = 512 threads = 16 waves: 2 M-tiles x 8 N-tiles.
// Row normalization is folded into the LDS staging (it commutes with the K-sum).
// In-place is safe: the block stages all rows it owns before writing them back.
__global__ __launch_bounds__(512) void k_gemm_hrs(float* __restrict__ S,
                                                  const float* __restrict__ weight,
                                                  const float* __restrict__ deg) {
    __shared__ float As[32 * FIN];          // 16 KB staged, pre-normalized A tile
    const int r     = blockIdx.y;
    const int mbase = blockIdx.x * 32;
    const int tid   = threadIdx.x;

    for (int i = tid; i < 32 * (FIN / 4); i += 512) {
        int rr = i >> 5, cc = i & 31;       // 32 float4 per row
        int grow = mbase + rr; if (grow > NN - 1) grow = NN - 1;
        float inv = 1.0f / (1.0f + deg[(size_t)r * NN + grow]);
        float4 v = ((const float4*)(S + ((size_t)r * NN + grow) * FIN))[cc];
        v.x *= inv; v.y *= inv; v.z *= inv; v.w *= inv;
        ((float4*)As)[i] = v;
    }
    __syncthreads();

    const int wave = tid >> 5, lane = tid & 31;
    const int mt = wave >> 3, nt = wave & 7;
    const int hi = lane >> 4, ln = lane & 15;
    const int arow = mt * 16 + ln;
    const int ncol = nt * 16 + ln;
    const float* wr = weight + (size_t)r * FIN * FOUT;

    v8f c = {};
#pragma unroll 8
    for (int ks = 0; ks < FIN / 4; ++ks) {
        int col = ks * 4 + 2 * hi;          // K {0,1} lanes 0-15, {2,3} lanes 16-31
        v2f a, b;
        a.x = As[arow * FIN + col];
        a.y = As[arow * FIN + col + 1];
        b.x = wr[(size_t)col * FOUT + ncol];
        b.y = wr[(size_t)(col + 1) * FOUT + ncol];
        c = __builtin_amdgcn_wmma_f32_16x16x4_f32(false, a, false, b, (short)0, c,
                                                  false, false);
    }

    float* outp = S + (size_t)r * NN * FOUT;    // in-place
    const int rbase = mbase + mt * 16 + 8 * hi;
    if (mbase + 32 <= NN) {                     // uniform fast path (no exec juggling)
#pragma unroll
        for (int i = 0; i < 8; ++i)
            outp[(size_t)(rbase + i) * FOUT + ncol] = c[i];
    } else {
#pragma unroll
        for (int i = 0; i < 8; ++i)
            if (rbase + i < NN) outp[(size_t)(rbase + i) * FOUT + ncol] = c[i];
    }
}

// ---------------- attention: u[n] = [Sum_r op*s1, Sum_r op*s2] ----------------
// one wave per node (4 channels / lane), softmax over R relations
__global__ __launch_bounds__(256) void k_attn(const float* __restrict__ op,
                                              const float* __restrict__ att1,
                                              const float* __restrict__ att2,
                                              float* __restrict__ u) {
    const int lane = threadIdx.x & 31;
    const int n    = blockIdx.x * 8 + (threadIdx.x >> 5);
    if (n >= NN) return;

    float4 a1 = ((const float4*)att1)[lane];
    float4 a2 = ((const float4*)att2)[lane];
    float4 opv[RR];
    float  e1[RR], e2[RR];

#pragma unroll
    for (int r = 0; r < RR; ++r) {
        float4 v = ((const float4*)(op + ((size_t)r * NN + n) * FOUT))[lane];
        opv[r] = v;
        float p1 = v.x * a1.x + v.y * a1.y + v.z * a1.z + v.w * a1.w;
        float p2 = v.x * a2.x + v.y * a2.y + v.z * a2.z + v.w * a2.w;
#pragma unroll
        for (int off = 16; off > 0; off >>= 1) {   // wave32 reduction
            p1 += __shfl_xor(p1, off, 32);
            p2 += __shfl_xor(p2, off, 32);
        }
        e1[r] = p1; e2[r] = p2;
    }

    float m1 = e1[0], m2 = e2[0];
#pragma unroll
    for (int r = 1; r < RR; ++r) { m1 = fmaxf(m1, e1[r]); m2 = fmaxf(m2, e2[r]); }
    float s1[RR], s2[RR], t1 = 0.f, t2 = 0.f;
#pragma unroll
    for (int r = 0; r < RR; ++r) {
        s1[r] = __expf(e1[r] - m1); t1 += s1[r];
        s2[r] = __expf(e2[r] - m2); t2 += s2[r];
    }
    float i1 = 1.0f / t1, i2 = 1.0f / t2;

    float4 u1 = make_float4(0.f, 0.f, 0.f, 0.f);
    float4 u2 = make_float4(0.f, 0.f, 0.f, 0.f);
#pragma unroll
    for (int r = 0; r < RR; ++r) {
        float w1 = s1[r] * i1, w2 = s2[r] * i2;
        u1.x += opv[r].x * w1; u1.y += opv[r].y * w1;
        u1.z += opv[r].z * w1; u1.w += opv[r].w * w1;
        u2.x += opv[r].x * w2; u2.y += opv[r].y * w2;
        u2.z += opv[r].z * w2; u2.w += opv[r].w * w2;
    }
    ((float4*)(u + (size_t)n * 256))[lane]       = u1;
    ((float4*)(u + (size_t)n * 256 + 128))[lane] = u2;
}

// ---------------- GEMM2: h = u[N,256] @ Wc^T + R*bc  (FP32 WMMA 16x16x4) ----------------
__global__ __launch_bounds__(512) void k_gemm_out(const float* __restrict__ u,
                                                  const float* __restrict__ Wc,
                                                  const float* __restrict__ bc,
                                                  float* __restrict__ out) {
    __shared__ float Us[32 * 256];          // 32 KB staged A tile
    const int mbase = blockIdx.x * 32;
    const int tid   = threadIdx.x;

    for (int i = tid; i < 32 * 64; i += 512) {
        int rr = i >> 6, cc = i & 63;       // 64 float4 per row
        int grow = mbase + rr; if (grow > NN - 1) grow = NN - 1;
        ((float4*)Us)[rr * 64 + cc] = ((const float4*)(u + (size_t)grow * 256))[cc];
    }
    __syncthreads();

    const int wave = tid >> 5, lane = tid & 31;
    const int mt = wave >> 3, nt = wave & 7;
    const int hi = lane >> 4, ln = lane & 15;
    const int arow = mt * 16 + ln;
    const int ncol = nt * 16 + ln;

    v8f c = {};
#pragma unroll 8
    for (int ks = 0; ks < 256 / 4; ++ks) {
        int col = ks * 4 + 2 * hi;
        v2f a;
        a.x = Us[arow * 256 + col];
        a.y = Us[arow * 256 + col + 1];
        // B[k,n] = Wc[n,k]: contiguous float2 per lane (Wc row-major [128,256])
        v2f b = *(const v2f*)(Wc + (size_t)ncol * 256 + col);
        c = __builtin_amdgcn_wmma_f32_16x16x4_f32(false, a, false, b, (short)0, c,
                                                  false, false);
    }

    float bias = (float)RR * bc[ncol];      // Sum over R of bc
    const int rbase = mbase + mt * 16 + 8 * hi;
    if (mbase + 32 <= NN) {                 // uniform fast path
#pragma unroll
        for (int i = 0; i < 8; ++i)
            out[(size_t)(rbase + i) * FOUT + ncol] = c[i] + bias;
    } else {
#pragma unroll
        for (int i = 0; i < 8; ++i)
            if (rbase + i < NN) out[(size_t)(rbase + i) * FOUT + ncol] = c[i] + bias;
    }
}

extern "C" void kernel_launch(void* const* d_in, const int* in_sizes, int n_in,
                              void* d_out, int out_size, void* d_ws, size_t ws_size,
                              hipStream_t stream) {
    const float* node_h = (const float*)d_in[0];
    const float* weight = (const float*)d_in[1];
    const float* att1   = (const float*)d_in[2];
    const float* att2   = (const float*)d_in[3];
    const float* Wc     = (const float*)d_in[4];
    const float* bc     = (const float*)d_in[5];
    const int*   esrc   = (const int*)d_in[6];
    const int*   edst   = (const int*)d_in[7];
    const int*   erel   = (const int*)d_in[8];
    float* out = (float*)d_out;
    float* ws  = (float*)d_ws;

    float* deg = ws + DEG_OFF;
    float* S   = ws + S_OFF;                // scatter target, then in-place op
    float* u   = ws + U_OFF;

    // zero deg + S (contiguous region)
    long n4 = (long)(DEG_SZ + S_SZ) / 4;
    k_zero<<<4096, 256, 0, stream>>>((float4*)ws, n4);

    k_deg<<<(EE + 255) / 256, 256, 0, stream>>>(erel, edst, deg);

    long scat_threads = (long)EE * 32;
    k_scatter<<<(int)((scat_threads + 255) / 256), 256, 0, stream>>>(node_h, esrc,
                                                                     edst, erel, S);

    dim3 g1((NN + 31) / 32, RR);
    k_gemm_hrs<<<g1, 512, 0, stream>>>(S, weight, deg);

    k_attn<<<(NN + 7) / 8, 256, 0, stream>>>(S, att1, att2, u);

    k_gemm_out<<<(NN + 31) / 32, 512, 0, stream>>>(u, Wc, bc, out);

    // second output of the reference tuple: weight passed through unchanged
    hipMemcpyAsync(out + (size_t)NN * FOUT, weight,
                   (size_t)RR * FIN * FOUT * sizeof(float),
                   hipMemcpyDeviceToDevice, stream);
}